// MatrixPrompt_51213190037813
// MI455X (gfx1250) — compile-verified
//
#include <hip/hip_runtime.h>
#include <hip/hip_bf16.h>
#include <math.h>

typedef __attribute__((ext_vector_type(2))) float v2f;
typedef __attribute__((ext_vector_type(4))) float v4f;
typedef __attribute__((ext_vector_type(8))) float v8f;

#define POOL   100
#define P_LEN  8
#define EMB_D  768
#define BATCH  512
#define TASK_END 10                 // (task_id+1) * POOL/N_TASKS with task_id==0
#define NFLAT  (P_LEN * EMB_D)      // 6144
#define KEY_SZ (BATCH * (P_LEN/2) * EMB_D)   // 1,572,864 floats

// ---------------------------------------------------------------------------
// Kernel 0: diag[k,d] = sum_l ep[k,l,d]^2      (10 x 768)
// ---------------------------------------------------------------------------
__global__ void diag_kernel(const float* __restrict__ ep, float* __restrict__ diag) {
    int i = blockIdx.x * blockDim.x + threadIdx.x;   // 0..7679
    if (i < TASK_END * EMB_D) {
        int k = i / EMB_D, d = i % EMB_D;
        float s = 0.f;
        #pragma unroll
        for (int l = 0; l < P_LEN; ++l) {
            float v = ep[k * NFLAT + l * EMB_D + d];
            s += v * v;
        }
        diag[i] = s;
    }
}

// ---------------------------------------------------------------------------
// Kernel 1: fused sim[b,k].  One 256-thread block per (b,k) pair.
// sim = sum(aq^2*diag) / (max(||aq*diag||,eps) * max(||aq||,eps))
// aq = x_querry[b, 90+k, :] * e_a[k, :]
// ---------------------------------------------------------------------------
__global__ void sim_kernel(const float* __restrict__ xq_base,
                           const float* __restrict__ e_a,
                           const float* __restrict__ diag,
                           float* __restrict__ sim) {
    int b = blockIdx.x / TASK_END;
    int k = blockIdx.x % TASK_END;
    const float* xq = xq_base + ((long)b * POOL + (POOL - TASK_END) + k) * EMB_D;
    const float* ea = e_a + k * EMB_D;
    const float* dg = diag + k * EMB_D;

    float s1 = 0.f, s2 = 0.f, s3 = 0.f;
    for (int d = threadIdx.x; d < EMB_D; d += 256) {
        float aq = xq[d] * ea[d];
        float ad = aq * dg[d];
        s1 += ad * aq;   // aq^2 * diag
        s2 += ad * ad;   // (aq*diag)^2
        s3 += aq * aq;   // aq^2
    }
    // intra-wave reduction (wave32)
    #pragma unroll
    for (int off = 16; off > 0; off >>= 1) {
        s1 += __shfl_xor(s1, off, 32);
        s2 += __shfl_xor(s2, off, 32);
        s3 += __shfl_xor(s3, off, 32);
    }
    __shared__ float r1[8], r2[8], r3[8];
    int lane = threadIdx.x & 31, wid = threadIdx.x >> 5;
    if (lane == 0) { r1[wid] = s1; r2[wid] = s2; r3[wid] = s3; }
    __syncthreads();
    if (threadIdx.x == 0) {
        float t1 = 0.f, t2 = 0.f, t3 = 0.f;
        #pragma unroll
        for (int w = 0; w < 8; ++w) { t1 += r1[w]; t2 += r2[w]; t3 += r3[w]; }
        float n1 = fmaxf(sqrtf(t2), 1e-12f);
        float n2 = fmaxf(sqrtf(t3), 1e-12f);
        sim[b * TASK_END + k] = t1 / (n1 * n2);
    }
}

// ---------------------------------------------------------------------------
// Kernel 2: int_pmt = sim(512x10) @ ep_flat(10x6144) via V_WMMA_F32_16X16X4_F32
// One wave per 16x16 output tile; K padded 10 -> 12 (3 WMMA issues).
// Tail chunk uses clamped addresses + v_cndmask zeroing: no EXEC divergence,
// wave stays convergent through all WMMAs.
// ---------------------------------------------------------------------------
__global__ void wmma_pmt_kernel(const float* __restrict__ sim,
                                const float* __restrict__ ep,
                                float* __restrict__ out) {
    int wave = (blockIdx.x * blockDim.x + threadIdx.x) >> 5;   // 0..12287
    int lane = threadIdx.x & 31;
    int tile_m = wave & 31;          // 32 M-tiles (batch/16)
    int tile_n = wave >> 5;          // 384 N-tiles (6144/16)
    int b0 = tile_m << 4;
    int n0 = tile_n << 4;
    int m    = lane & 15;            // row of A / column of B and D
    int half = lane >> 4;            // 0 or 1

    v8f c = {};
    #pragma unroll
    for (int kb = 0; kb < 12; kb += 4) {
        int k0 = kb + 2 * half;
        int k1 = k0 + 1;
        // clamped (always in-bounds) addresses -> unconditional loads
        int k0c = (k0 < TASK_END) ? k0 : (TASK_END - 1);
        int k1c = (k1 < TASK_END) ? k1 : (TASK_END - 1);
        float a0 = sim[(b0 + m) * TASK_END + k0c];
        float a1 = sim[(b0 + m) * TASK_END + k1c];
        float e0 = ep[k0c * NFLAT + n0 + m];
        float e1 = ep[k1c * NFLAT + n0 + m];
        v2f a, b;
        a.x = (k0 < TASK_END) ? a0 : 0.f;   // v_cndmask, no branch
        a.y = (k1 < TASK_END) ? a1 : 0.f;
        b.x = (k0 < TASK_END) ? e0 : 0.f;
        b.y = (k1 < TASK_END) ? e1 : 0.f;
        c = __builtin_amdgcn_wmma_f32_16x16x4_f32(
                /*neg_a=*/false, a, /*neg_b=*/false, b,
                /*c_mod=*/(short)0, c, /*reuse_a=*/false, /*reuse_b=*/false);
    }

    int l = n0 / EMB_D;              // prompt row (wave-uniform: 16 | 768)
    int d = n0 % EMB_D + m;          // column within row
    float* base = (l < P_LEN/2)
        ? out           + l              * EMB_D
        : out + KEY_SZ  + (l - P_LEN/2) * EMB_D;
    #pragma unroll
    for (int i = 0; i < 8; ++i) {
        int bb = b0 + i + 8 * half;
        base[(long)bb * ((P_LEN/2) * EMB_D) + d] = c[i];
    }
}

// ---------------------------------------------------------------------------
// Kernel 3: x_block pass-through copy (dominant cost, pure bandwidth).
// Non-temporal load+store: 310MB stream must not thrash the 192MB L2.
// Native clang vector type (v4f) so the nontemporal builtins accept it.
// ---------------------------------------------------------------------------
__global__ void copy_kernel(const v4f* __restrict__ src,
                            v4f* __restrict__ dst, long n4) {
    long i = (long)blockIdx.x * blockDim.x + threadIdx.x;
    if (i < n4) {
        v4f v = __builtin_nontemporal_load(&src[i]);
        __builtin_nontemporal_store(v, &dst[i]);
    }
}

// ---------------------------------------------------------------------------
extern "C" void kernel_launch(void* const* d_in, const int* in_sizes, int n_in,
                              void* d_out, int out_size, void* d_ws, size_t ws_size,
                              hipStream_t stream) {
    const float* x_querry = (const float*)d_in[0];   // (512,100,768)
    const float* x_block  = (const float*)d_in[1];   // (512,197,768)
    const float* e_a      = (const float*)d_in[2];   // (100,768)
    const float* e_p      = (const float*)d_in[3];   // (100,8,768)
    float* out = (float*)d_out;

    float* diag = (float*)d_ws;                 // 10*768 floats
    float* sim  = diag + TASK_END * EMB_D;      // 512*10 floats

    // diag: 7680 elems
    diag_kernel<<<(TASK_END * EMB_D + 255) / 256, 256, 0, stream>>>(e_p, diag);

    // sim: one block per (b,k)
    sim_kernel<<<BATCH * TASK_END, 256, 0, stream>>>(x_querry, e_a, diag, sim);

    // WMMA GEMM: 32*384 = 12288 tiles, 8 waves/block -> 1536 blocks
    wmma_pmt_kernel<<<(32 * 384) / 8, 256, 0, stream>>>(sim, e_p, out);

    // x_block copy into out[2*KEY_SZ ...]
    long n4 = ((long)BATCH * 197 * EMB_D) / 4;
    copy_kernel<<<(unsigned)((n4 + 255) / 256), 256, 0, stream>>>(
        (const v4f*)x_block, (v4f*)(out + 2 * (long)KEY_SZ), n4);
}